// ROIPooler_73804718015059
// MI455X (gfx1250) — compile-verified
//
#include <hip/hip_runtime.h>
#include <stdint.h>

#define RP 7        // pooled resolution
#define NC 256      // channels

typedef unsigned int v4u __attribute__((ext_vector_type(4)));
typedef int          v8i __attribute__((ext_vector_type(8)));
typedef int          v4i __attribute__((ext_vector_type(4)));
typedef float        vf2 __attribute__((ext_vector_type(2)));
typedef vf2 __attribute__((aligned(4))) vf2_u;   // 4B-aligned pair load -> b64

// ---------------------------------------------------------------------------
// Kernel 1: per-ROI sampling tables -> workspace (128 floats / ROI, 512B)
//   y slots: [0..13] ylo*L (int bits)  [14..27] yhi*L  [28..41] wy  [42..55] vy
//   x slots: [56..69] xp=min(xlo,L-2)  [70..83] sel    [84..97] wx  [98..111] vx
//   [112] level (int bits)   [113] batch index (int bits)
// ---------------------------------------------------------------------------
__global__ __launch_bounds__(32)
void roi_table_kernel(const float* __restrict__ boxes,
                      const int*   __restrict__ bidx,
                      float*       __restrict__ tab) {
  const int r = blockIdx.x;
  const int t = threadIdx.x;
  const float b0 = boxes[r * 4 + 0], b1 = boxes[r * 4 + 1];
  const float b2 = boxes[r * 4 + 2], b3 = boxes[r * 4 + 3];

  // level assignment (matches _assign_levels)
  const float bw = fmaxf(b2 - b0, 0.0f), bh = fmaxf(b3 - b1, 0.0f);
  const float sz = sqrtf(bw * bh);
  const float lf = floorf(4.0f + log2f(sz / 224.0f + 1e-8f));
  const int   lvl = (int)fminf(fmaxf(lf, 2.0f), 5.0f) - 2;   // 0..3
  const float scales[4] = {0.25f, 0.125f, 0.0625f, 0.03125f};
  const float scale = scales[lvl];
  const int   L = 256 >> lvl;                                // H == W per level

  float* row = tab + (size_t)r * 128;

  if (t < 28) {
    const bool  isx   = (t >= 14);
    const int   j     = isx ? (t - 14) : t;                  // sample index 0..13
    const float start = (isx ? b0 : b1) * scale - 0.5f;
    const float binsz = ((isx ? (b2 - b0) : (b3 - b1)) * scale) * (1.0f / 7.0f);
    const float off   = (float)(j >> 1) + ((j & 1) ? 0.75f : 0.25f);
    const float cpos  = start + binsz * off;
    const float valid = (cpos >= -1.0f && cpos <= (float)L) ? 1.0f : 0.0f;
    const float tc    = fminf(fmaxf(cpos, 0.0f), (float)(L - 1));
    const int   lo    = (int)floorf(tc);
    const float frac  = tc - (float)lo;
    if (isx) {
      const int   xp  = (lo < L - 1) ? lo : L - 2;  // pair base for b64 load
      const float sel = (lo < L - 1) ? 1.0f : 0.0f; // 1 -> lo value is pair.x
      row[56 + j] = __int_as_float(xp);
      row[70 + j] = sel;
      row[84 + j] = frac;
      row[98 + j] = valid;
    } else {
      const int hi = (lo + 1 < L) ? lo + 1 : L - 1;
      row[0  + j] = __int_as_float(lo * L);         // premultiplied row offset
      row[14 + j] = __int_as_float(hi * L);
      row[28 + j] = frac;
      row[42 + j] = valid;
    }
  } else if (t == 28) {
    row[112] = __int_as_float(lvl);
  } else if (t == 29) {
    row[113] = __int_as_float(bidx[r]);
  }
}

// ---------------------------------------------------------------------------
// Kernel 2: main ROIAlign. grid = (R, 8 channel-groups), 256 threads (8 waves).
// Table staged to LDS via the Tensor Data Mover (D# per CDNA5 ISA).
// ---------------------------------------------------------------------------
__global__ __launch_bounds__(256)
void roi_align_kernel(const float* __restrict__ f2, const float* __restrict__ f3,
                      const float* __restrict__ f4, const float* __restrict__ f5,
                      const float* __restrict__ tab, float* __restrict__ out) {
  __shared__ float lt[128];

  const int r  = blockIdx.x;
  const int cg = blockIdx.y;

  if (threadIdx.x < 32) {  // wave 0 issues the TDM descriptor
    const uint32_t lds_off = (uint32_t)(uintptr_t)(&lt[0]);        // low 32b = LDS byte addr
    const uint64_t ga      = (uint64_t)(uintptr_t)(tab + (size_t)r * 128);

    // D# group 0: count=1 | lds_addr | global_addr[56:0] | type=2 ("image")
    v4u G0 = { 1u,
               lds_off,
               (uint32_t)ga,
               (uint32_t)((ga >> 32) & 0x1FFFFFFull) | 0x80000000u };
    // D# group 1: wg_mask=0, data_size=2 (4B), tensor_dim0=128, tensor_dim1=1,
    //             tile_dim0=128, tile_dim1=1, tensor_dim0_stride=128
    v8i G1 = { (int)(2u   << 16),   // data_size = 4B
               (int)(128u << 16),   // tensor_dim0 lo
               (int)(1u   << 16),   // tensor_dim1 lo
               (int)(128u << 16),   // tile_dim0
               1,                   // tile_dim1 = 1, tile_dim2 = 0
               128,                 // tensor_dim0_stride lo
               0, 0 };
    v4i Z4 = {0, 0, 0, 0};
    v8i Z8 = {0, 0, 0, 0, 0, 0, 0, 0};
    __builtin_amdgcn_tensor_load_to_lds(G0, G1, Z4, Z4, Z8, 0);
    __builtin_amdgcn_s_wait_tensorcnt(0);
  }
  __syncthreads();

  const int lvl = __float_as_int(lt[112]);
  const int b   = __float_as_int(lt[113]);
  const float* feat = (lvl == 0) ? f2 : (lvl == 1) ? f3 : (lvl == 2) ? f4 : f5;
  const int L  = 256 >> lvl;
  const int HW = L * L;

  // 32 channels x 49 bins per block; bin fastest -> x-adjacent lanes
  for (int idx = threadIdx.x; idx < 32 * RP * RP; idx += 256) {
    const int cl  = idx / (RP * RP);
    const int bin = idx - cl * (RP * RP);
    const int py  = bin / RP;
    const int px  = bin - py * RP;
    const int c   = cg * 32 + cl;
    const float* base = feat + ((size_t)(b * NC + c)) * HW;

    // x-side prep (independent of sy): pair base, selector, lerp w, valid
    int   xp[2];
    float sel[2], wx[2], vx[2];
#pragma unroll
    for (int sx = 0; sx < 2; ++sx) {
      const int ix = px * 2 + sx;
      xp[sx]  = __float_as_int(lt[56 + ix]);
      sel[sx] = lt[70 + ix];
      wx[sx]  = lt[84 + ix];
      vx[sx]  = lt[98 + ix];
    }

    float acc = 0.0f;
#pragma unroll
    for (int sy = 0; sy < 2; ++sy) {
      const int   iy  = py * 2 + sy;
      const int   ro0 = __float_as_int(lt[iy]);       // ylo*L
      const int   ro1 = __float_as_int(lt[14 + iy]);  // yhi*L
      const float wy  = lt[28 + iy];
      const float vy  = lt[42 + iy];
      const float* r0 = base + ro0;
      const float* r1 = base + ro1;
#pragma unroll
      for (int sx = 0; sx < 2; ++sx) {
        // one b64 per row fetches both x-corners (xhi = xp+1 always)
        const vf2 a = *(const vf2_u*)(r0 + xp[sx]);
        const vf2 b2v = *(const vf2_u*)(r1 + xp[sx]);
        const float a_lo = fmaf(sel[sx], a.x - a.y, a.y);     // sel ? a.x : a.y
        const float b_lo = fmaf(sel[sx], b2v.x - b2v.y, b2v.y);
        const float a_v  = a_lo + wx[sx] * (a.y - a_lo);      // x-lerp, top row
        const float b_v  = b_lo + wx[sx] * (b2v.y - b_lo);    // x-lerp, bottom row
        acc += (vy * vx[sx]) * (a_v + wy * (b_v - a_v));      // y-lerp + mask
      }
    }
    // one-time streamed output: non-temporal so it doesn't evict the
    // feature pyramid (178MB) from the 192MB L2
    const size_t o = ((size_t)r * NC + c) * (RP * RP) + bin;
    __builtin_nontemporal_store(acc * 0.25f, out + o);
  }
}

extern "C" void kernel_launch(void* const* d_in, const int* in_sizes, int n_in,
                              void* d_out, int out_size, void* d_ws, size_t ws_size,
                              hipStream_t stream) {
  const float* f2    = (const float*)d_in[0];
  const float* f3    = (const float*)d_in[1];
  const float* f4    = (const float*)d_in[2];
  const float* f5    = (const float*)d_in[3];
  const float* boxes = (const float*)d_in[4];
  const int*   bidx  = (const int*)d_in[5];
  float*       tab   = (float*)d_ws;            // R * 128 floats (512B/ROI)
  const int    R     = in_sizes[4] / 4;

  roi_table_kernel<<<dim3(R), dim3(32), 0, stream>>>(boxes, bidx, tab);
  roi_align_kernel<<<dim3(R, 8), dim3(256), 0, stream>>>(f2, f3, f4, f5, tab,
                                                         (float*)d_out);
}